// HierarchicalSoftmaxPyTorch_90254442758228
// MI455X (gfx1250) — compile-verified
//
#include <hip/hip_runtime.h>

// Problem sizes (match reference)
#define B_SZ 16384
#define D_SZ 128
#define L_SZ 24
#define N_SZ 100000

typedef __attribute__((ext_vector_type(2))) float v2f;
typedef __attribute__((ext_vector_type(8))) float v8f;

// One wave32 handles a tile of 16 batch rows.
// A (16x4 f32 per k-step): input_vectors rows, preloaded once, reused for all 24 levels.
// B (4x16 f32 per k-step): gathered embedding rows for the 16 pairs at level l.
// C (16x16 f32): accumulated over 32 k-steps; diagonal C[i,i] = dot(iv[b0+i], emb[p_i]).
__global__ __launch_bounds__(256) void hs_loss_wmma_kernel(
    const float* __restrict__ iv,      // [B, D]
    const float* __restrict__ emb,     // [N, D]
    const int*   __restrict__ codes,   // [B, L]
    const int*   __restrict__ points,  // [B, L]
    const int*   __restrict__ lengths, // [B]
    float* __restrict__ out)           // [1]
{
    const int lane = threadIdx.x & 31;
    const int wave = threadIdx.x >> 5;
    const int tile = blockIdx.x * 8 + wave;   // 128 blocks * 8 waves = 1024 tiles
    const int b0   = tile * 16;

    const int n  = lane & 15;   // matrix column (pair index within tile) / A row
    const int hi = lane >> 4;   // 0: K={0,1} of each group of 4; 1: K={2,3}
    const int brow = b0 + n;

    // ---- Preload A fragments (full D=128 row split across lane pairs) ----
    // ISA A-layout (32-bit 16x4): lanes 0-15 hold M=0..15 with K=0,1 in v0.x,v1.x;
    // lanes 16-31 hold the same M with K=2,3.  Per k-step kk, lane reads
    // iv[brow, 4*kk + 2*hi + {0,1}] as one float2.
    v2f a[32];
    const float* iv_row = iv + (size_t)brow * D_SZ + 2 * hi;
#pragma unroll
    for (int kk = 0; kk < 32; ++kk) {
        a[kk] = *(const v2f*)(iv_row + 4 * kk);
    }

    const int len_n = lengths[brow];

    float lsum = 0.0f;

    for (int l = 0; l < L_SZ; ++l) {
        // Gather index + validity for pair n at level l (branchless, EXEC stays full)
        int  p     = points[brow * L_SZ + l];
        bool valid = (l < len_n) & (p >= 0) & (p < N_SZ);
        int  pc    = valid ? p : 0;                     // clamp to keep gather in-bounds
        const float* erow = emb + (size_t)pc * D_SZ + 2 * hi;

        // ---- 32 x V_WMMA_F32_16X16X4_F32 accumulation (K = 128) ----
        v8f c = {0.f, 0.f, 0.f, 0.f, 0.f, 0.f, 0.f, 0.f};
#pragma unroll
        for (int kk = 0; kk < 32; ++kk) {
            v2f bfrag = *(const v2f*)(erow + 4 * kk);   // B-layout mirrors A (transposed)
            c = __builtin_amdgcn_wmma_f32_16x16x4_f32(
                    /*neg_a=*/false, a[kk],
                    /*neg_b=*/false, bfrag,
                    /*c_mod=*/(short)0, c,
                    /*reuse_a=*/false, /*reuse_b=*/false);
        }

        // ---- Diagonal extraction ----
        // C layout: lane holds N=lane&15; VGPR r holds M = r + 8*(lane>=16).
        // Diag owners: lanes 0..7 (pair=lane, vgpr=lane) and lanes 24..31
        // (pair=lane-16, vgpr=lane-24).  In both cases vgpr = lane&7 and the
        // owned pair equals this lane's column n.
        const int idx = lane & 7;
        float dot = c[0];
        dot = (idx == 1) ? c[1] : dot;
        dot = (idx == 2) ? c[2] : dot;
        dot = (idx == 3) ? c[3] : dot;
        dot = (idx == 4) ? c[4] : dot;
        dot = (idx == 5) ? c[5] : dot;
        dot = (idx == 6) ? c[6] : dot;
        dot = (idx == 7) ? c[7] : dot;
        const bool owner = (lane < 8) | (lane >= 24);

        // ---- NLL epilogue (fp32, matches reference formula incl. eps) ----
        float cd  = (float)codes[brow * L_SZ + l];
        float sig = 1.0f / (1.0f + __expf(-dot));
        float nll = -(cd * __logf(sig + 1e-10f) +
                      (1.0f - cd) * __logf(1.0f - sig + 1e-10f));
        lsum += (owner && valid) ? nll : 0.0f;
    }

    // ---- Wave32 butterfly reduction, one global atomic per wave ----
#pragma unroll
    for (int off = 16; off > 0; off >>= 1)
        lsum += __shfl_xor(lsum, off, 32);

    if (lane == 0)
        unsafeAtomicAdd(out, lsum * (1.0f / (float)B_SZ));
}

extern "C" void kernel_launch(void* const* d_in, const int* in_sizes, int n_in,
                              void* d_out, int out_size, void* d_ws, size_t ws_size,
                              hipStream_t stream) {
    const float* iv      = (const float*)d_in[0];
    const float* emb     = (const float*)d_in[1];
    const int*   codes   = (const int*)d_in[2];
    const int*   points  = (const int*)d_in[3];
    const int*   lengths = (const int*)d_in[4];
    float*       out     = (float*)d_out;

    // d_out is poisoned by the harness; zero it (memset node is graph-capturable).
    hipMemsetAsync(out, 0, out_size * sizeof(float), stream);

    // 1024 tiles of 16 batch rows; 8 waves (tiles) per 256-thread block.
    dim3 grid(B_SZ / (16 * 8));
    dim3 block(256);
    hs_loss_wmma_kernel<<<grid, block, 0, stream>>>(iv, emb, codes, points, lengths, out);
}